// MartiniBackmapper_18425409699909
// MI455X (gfx1250) — compile-verified
//
#include <hip/hip_runtime.h>
#include <hip/hip_bf16.h>

#ifndef INV_SQRT2_F
#define INV_SQRT2_F 0.70710678118654752440f
#endif

#define BGROUP 8

// ---------------------------------------------------------------------------
// Never-launched probe kernel, placed FIRST in .text so the disasm snippet
// shows exactly how the three key constructs lower on gfx1250:
//   1) atomicAdd(float*)           -> expect native global_atomic_add_f32
//   2) __builtin_nontemporal_store -> expect global_store_b32 with NT hint
//   3) __builtin_prefetch          -> expect global_prefetch_b8
// ---------------------------------------------------------------------------
__global__ void aa_probe_kernel(float* __restrict__ p, float v) {
  float* q = p + threadIdx.x;
  atomicAdd(q, v);                         // (1)
  __builtin_nontemporal_store(v, q + 64);  // (2)
  __builtin_prefetch(q + 128, 0, 0);       // (3)
}

// ---------------------------------------------------------------------------
// Main backmapping kernel.
//   grid.x covers m in [0, M), grid.y covers batch groups of BGROUP b's.
// Per-m work (indices, weights, sincos, scaling) hoisted out of the b-loop.
// rot_flat written with non-temporal (streaming) stores; scatter uses native
// f32 global atomics (result ignored -> no-return atomic).
// ---------------------------------------------------------------------------
__global__ __launch_bounds__(256) void mb_backmap_kernel(
    const float* __restrict__ P,       // (B, N_BEADS, 3)
    const float* __restrict__ p1,      // (B, N_BEADS, 3)
    const float* __restrict__ p2,      // (B, N_BEADS, 3)
    const float* __restrict__ Wt,      // (3, N_BEADS, BEAD_SIZE)
    const float* __restrict__ ang,     // (M, 1)
    const float* __restrict__ scal,    // (N_ATOMS, 1)
    const int*   __restrict__ slot_idx,// (M,)
    const int*   __restrict__ atom_idx,// (M,)
    const int*   __restrict__ bead_idx,// (M,)
    float* __restrict__ out,           // (B, N_ATOMS, 3)  -- atomic accum
    float* __restrict__ rot_out,       // (B, M, 3)        -- streamed
    int M, int B, int nb3, int wstride, int na3) {
  int m = blockIdx.x * blockDim.x + threadIdx.x;
  if (m >= M) return;

  int b0   = (int)blockIdx.y * BGROUP;
  int bend = b0 + BGROUP;
  if (bend > B) bend = B;

  // ---- per-m (batch-invariant) work, hoisted out of the b-loop ----
  const int slot = slot_idx[m];
  const int atom = atom_idx[m];
  const int bead = bead_idx[m];

  const float w0 = Wt[slot];
  const float w1 = Wt[wstride + slot];
  const float w2 = Wt[2 * wstride + slot];

  float s, c;
  sincosf(ang[m], &s, &c);
  const float sc = scal[atom];

  const float* p1p = p1 + (long long)b0 * nb3 + (long long)bead * 3;
  const float* p2p = p2 + (long long)b0 * nb3 + (long long)bead * 3;
  const float* Pp  = P  + (long long)b0 * nb3 + (long long)bead * 3;
  float* rp = rot_out + ((long long)b0 * M + m) * 3;
  float* op = out + (long long)b0 * na3 + (long long)atom * 3;

  const long long rstep = (long long)M * 3;

#pragma unroll 2
  for (int b = b0; b < bend; ++b) {
    if (b + 1 < bend) {  // gfx1250 global_prefetch for next strided rows
      __builtin_prefetch(p1p + nb3, 0, 0);
      __builtin_prefetch(p2p + nb3, 0, 0);
      __builtin_prefetch(Pp + nb3, 0, 0);
    }

    const float x1 = p1p[0], y1 = p1p[1], z1 = p1p[2];
    const float x2 = p2p[0], y2 = p2p[1], z2 = p2p[2];

    // p12 = cross(p1, p2) * 1/sqrt(2)
    const float cx = (y1 * z2 - z1 * y2) * INV_SQRT2_F;
    const float cy = (z1 * x2 - x1 * z2) * INV_SQRT2_F;
    const float cz = (x1 * y2 - y1 * x2) * INV_SQRT2_F;

    // rot = w0*p1 + w1*p2 + w2*p12, normalized
    float rx = w0 * x1 + w1 * x2 + w2 * cx;
    float ry = w0 * y1 + w1 * y2 + w2 * cy;
    float rz = w0 * z1 + w1 * z2 + w2 * cz;
    const float inv = 1.0f / sqrtf(rx * rx + ry * ry + rz * rz);
    rx *= inv; ry *= inv; rz *= inv;

    // rot_flat output: write-once stream -> non-temporal stores
    __builtin_nontemporal_store(rx, rp + 0);
    __builtin_nontemporal_store(ry, rp + 1);
    __builtin_nontemporal_store(rz, rp + 2);

    // quaternion sandwich: q = (c, s*rot), v = p1[b, bead]
    const float qw = c, qx = s * rx, qy = s * ry, qz = s * rz;
    const float vx = x1, vy = y1, vz = z1;

    const float tw = -qx * vx - qy * vy - qz * vz;
    const float tx =  qw * vx + qy * vz - qz * vy;
    const float ty =  qw * vy + qz * vx - qx * vz;
    const float tz =  qw * vz + qx * vy - qy * vx;

    const float ox = -tw * qx + tx * qw - ty * qz + tz * qy;
    const float oy = -tw * qy + ty * qw - tz * qx + tx * qz;
    const float oz = -tw * qz + tz * qw - tx * qy + ty * qx;

    // recon = scaling * v_rot + P ; scatter-add into out[b, atom]
    atomicAdd(op + 0, sc * ox + Pp[0]);
    atomicAdd(op + 1, sc * oy + Pp[1]);
    atomicAdd(op + 2, sc * oz + Pp[2]);

    p1p += nb3; p2p += nb3; Pp += nb3;
    rp += rstep;
    op += na3;
  }
}

// ---------------------------------------------------------------------------
// Zero the scatter-accumulated output region (B * N_ATOMS * 3 floats).
// Vectorized float4 stores; fixed 4-iteration guarded tail for clean codegen.
// Runs every launch so the atomic accumulation starts from zero.
// ---------------------------------------------------------------------------
__global__ __launch_bounds__(256) void mb_zero_kernel(float* __restrict__ p,
                                                      long long n) {
  long long i = ((long long)blockIdx.x * blockDim.x + threadIdx.x) * 4;
  if (i + 3 < n) {
    *reinterpret_cast<float4*>(p + i) = make_float4(0.f, 0.f, 0.f, 0.f);
  } else {
#pragma unroll
    for (int k = 0; k < 4; ++k) {
      if (i + k < n) p[i + k] = 0.f;
    }
  }
}

// ---------------------------------------------------------------------------
// Host-side launcher. Input order (from setup_inputs):
//   0:P  1:p1  2:p2  3:weights  4:angles_polar  5:scaling
//   6:slot_idx  7:atom_idx  8:bead_idx
// d_out = [ out (B*N_ATOMS*3) | rot_flat (B*M*3) ]  (float32)
// ---------------------------------------------------------------------------
extern "C" void kernel_launch(void* const* d_in, const int* in_sizes, int n_in,
                              void* d_out, int out_size, void* d_ws, size_t ws_size,
                              hipStream_t stream) {
  (void)n_in; (void)d_ws; (void)ws_size;

  const float* P    = (const float*)d_in[0];
  const float* p1   = (const float*)d_in[1];
  const float* p2   = (const float*)d_in[2];
  const float* Wt   = (const float*)d_in[3];
  const float* ang  = (const float*)d_in[4];
  const float* scal = (const float*)d_in[5];
  const int* slot_idx = (const int*)d_in[6];
  const int* atom_idx = (const int*)d_in[7];
  const int* bead_idx = (const int*)d_in[8];

  const int M       = in_sizes[6];            // slot_idx count
  const int N_ATOMS = in_sizes[5];            // scaling rows
  const int B       = out_size / (3 * (N_ATOMS + M));
  const int N_BEADS = in_sizes[0] / (3 * B);  // P = (B, N_BEADS, 3)
  const int wstride = in_sizes[3] / 3;        // N_BEADS * BEAD_SIZE
  const int nb3     = 3 * N_BEADS;
  const int na3     = 3 * N_ATOMS;

  float* out_acc = (float*)d_out;
  float* rot_out = out_acc + (long long)B * na3;

  // 1) zero the atomic-accumulation region
  {
    const long long n = (long long)B * na3;
    const long long n4 = (n + 3) / 4;
    const int threads = 256;
    const long long blocks = (n4 + threads - 1) / threads;
    mb_zero_kernel<<<dim3((unsigned)blocks), dim3(threads), 0, stream>>>(out_acc, n);
  }

  // 2) main backmapping + scatter
  {
    const int threads = 256;
    const unsigned gx = (unsigned)((M + threads - 1) / threads);
    const unsigned gy = (unsigned)((B + BGROUP - 1) / BGROUP);
    mb_backmap_kernel<<<dim3(gx, gy), dim3(threads), 0, stream>>>(
        P, p1, p2, Wt, ang, scal, slot_idx, atom_idx, bead_idx,
        out_acc, rot_out, M, B, nb3, wstride, na3);
  }
}